// MultiScaleDeformableAttention_53609781788848
// MI455X (gfx1250) — compile-verified
//
#include <hip/hip_runtime.h>
#include <hip/hip_bf16.h>
#include <math.h>

// ---------------------------------------------------------------------------
// MultiScaleDeformableAttention for MI455X (gfx1250), fp32 end-to-end.
//   GEMMs via V_WMMA_F32_16X16X4_F32 (wave32 matrix pipe, f32 precision).
//   Wave tile = 32(M) x 64(N): 8 accumulators, B fragments reused across two
//   A fragments -> 8 WMMA per 10 VMEM per K-step (2x arithmetic intensity vs
//   a 16x64 tile).
//   Softmax fused into the attention-weight GEMM epilogue (shfl_xor within
//   16-lane half-wave == one head's 16 logits).
//   Sampling: one wave per (n,q,head); lane == channel -> coalesced gathers.
// ---------------------------------------------------------------------------

typedef __attribute__((ext_vector_type(2))) float v2f;
typedef __attribute__((ext_vector_type(8))) float v8f;

#define EMBED_DIM 256
#define N_HEADS   8
#define N_LEVELS  4
#define N_POINTS  4
#define HEAD_DIM  32
#define LV_TOTAL  13294   // 100*100 + 50*50 + 25*25 + 13*13

// ---------------------------------------------------------------------------
// WMMA GEMM:  C[M,Nn] = A[M,K] @ W[K,Nn] + bias[Nn]
// One wave computes a 32(M) x 64(N) strip: acc[mi][nj], mi in {0,1} (16-row
// halves), nj in {0..3} (16-col subtiles).
// SOFTMAX16 != 0: softmax over each aligned 16-column group per row in the
// epilogue (attention-weight projection, Nn=128; 16 cols == one head).
// ---------------------------------------------------------------------------
template <int SOFTMAX16>
__launch_bounds__(256)
__global__ void gemm_bias_wmma(const float* __restrict__ A,
                               const float* __restrict__ W,
                               const float* __restrict__ bias,
                               float* __restrict__ C,
                               int M, int K, int Nn)
{
    const int lane = threadIdx.x & 31;
    const int wave = threadIdx.x >> 5;

    const int tilesN4 = Nn >> 6;                  // 64-wide macro tiles
    const int tilesM  = (M + 31) >> 5;            // 32-row macro tiles
    const long tid = (long)blockIdx.x * (blockDim.x >> 5) + wave;
    if (tid >= (long)tilesM * tilesN4) return;    // whole-wave uniform exit
    const int tM = (int)(tid / tilesN4);
    const int tN = (int)(tid % tilesN4);

    // A fragment (16x4 f32): lane l<16 -> row l, K = k0 + {0,1};
    //                        lane l>=16 -> row l-16, K = k0 + {2,3}.
    const int rloc  = lane & 15;
    const int khalf = (lane >> 4) << 1;           // 0 or 2
    int row0 = tM * 32 + rloc;
    int row1 = tM * 32 + 16 + rloc;
    int row0c = row0 < M ? row0 : (M - 1);        // clamp tail (stores guarded)
    int row1c = row1 < M ? row1 : (M - 1);
    const float* Ap0 = A + (size_t)row0c * K + khalf;
    const float* Ap1 = A + (size_t)row1c * K + khalf;

    // B fragment (4x16 f32): lane l<16 -> col l, K = k0 + {0,1} in vgpr {0,1};
    //                        lane l>=16 -> col l-16, K = k0 + {2,3}.
    const int colb = tN * 64 + (lane & 15);
    const float* Bp = W + (size_t)khalf * Nn + colb;

    v8f acc[2][4] = {};
#pragma unroll 4
    for (int k0 = 0; k0 < K; k0 += 4) {
        v2f a0 = *(const v2f*)(Ap0 + k0);         // contiguous pair -> b64 load
        v2f a1 = *(const v2f*)(Ap1 + k0);
        const float* bp = Bp + (size_t)k0 * Nn;
        v2f b0 = { bp[0],  bp[(size_t)Nn + 0]  };
        v2f b1 = { bp[16], bp[(size_t)Nn + 16] };
        v2f b2 = { bp[32], bp[(size_t)Nn + 32] };
        v2f b3 = { bp[48], bp[(size_t)Nn + 48] };
        acc[0][0] = __builtin_amdgcn_wmma_f32_16x16x4_f32(false, a0, false, b0,
                                                          (short)0, acc[0][0], false, false);
        acc[0][1] = __builtin_amdgcn_wmma_f32_16x16x4_f32(false, a0, false, b1,
                                                          (short)0, acc[0][1], false, false);
        acc[0][2] = __builtin_amdgcn_wmma_f32_16x16x4_f32(false, a0, false, b2,
                                                          (short)0, acc[0][2], false, false);
        acc[0][3] = __builtin_amdgcn_wmma_f32_16x16x4_f32(false, a0, false, b3,
                                                          (short)0, acc[0][3], false, false);
        acc[1][0] = __builtin_amdgcn_wmma_f32_16x16x4_f32(false, a1, false, b0,
                                                          (short)0, acc[1][0], false, false);
        acc[1][1] = __builtin_amdgcn_wmma_f32_16x16x4_f32(false, a1, false, b1,
                                                          (short)0, acc[1][1], false, false);
        acc[1][2] = __builtin_amdgcn_wmma_f32_16x16x4_f32(false, a1, false, b2,
                                                          (short)0, acc[1][2], false, false);
        acc[1][3] = __builtin_amdgcn_wmma_f32_16x16x4_f32(false, a1, false, b3,
                                                          (short)0, acc[1][3], false, false);
    }

    // C fragment: VGPR r, lanes 0-15 -> row r cols 0-15;
    //             lanes 16-31 -> row r+8 cols 0-15.
    const int rowHi = (lane >> 4) << 3;           // 0 or 8
#pragma unroll
    for (int mi = 0; mi < 2; ++mi) {
#pragma unroll
        for (int j = 0; j < 4; ++j) {
            const int col = tN * 64 + j * 16 + (lane & 15);
            const float bv = bias[col];
#pragma unroll
            for (int r = 0; r < 8; ++r) {
                float v = acc[mi][j][r] + bv;
                if (SOFTMAX16) {
                    // softmax over the 16 columns of this row: xor masks 1..8
                    // stay inside each 16-lane half.
                    float m = v;
                    m = fmaxf(m, __shfl_xor(m, 8, 32));
                    m = fmaxf(m, __shfl_xor(m, 4, 32));
                    m = fmaxf(m, __shfl_xor(m, 2, 32));
                    m = fmaxf(m, __shfl_xor(m, 1, 32));
                    float e = __expf(v - m);
                    float s = e;
                    s += __shfl_xor(s, 8, 32);
                    s += __shfl_xor(s, 4, 32);
                    s += __shfl_xor(s, 2, 32);
                    s += __shfl_xor(s, 1, 32);
                    v = e / s;
                }
                const int orow = tM * 32 + mi * 16 + r + rowHi;
                if (orow < M) C[(size_t)orow * Nn + col] = v;
            }
        }
    }
}

// ---------------------------------------------------------------------------
// Deformable-attention sampling core.
// One wave handles one (n, q, head); lane == channel (HEAD_DIM == 32), so
// every corner gather is one coalesced 128B line from the projected value.
// All location/weight math is wave-uniform (SALU-friendly, no divergence).
// ---------------------------------------------------------------------------
__launch_bounds__(256)
__global__ void msda_sample_kernel(const float* __restrict__ val,   // (N,Lv,256)
                                   const float* __restrict__ off,   // (N,Lq,256)
                                   const float* __restrict__ aw,    // (N,Lq,128)
                                   const float* __restrict__ refp,  // (N,Lq,4,4)
                                   float* __restrict__ out,         // (N,Lq,256)
                                   int Lq, int Lv)
{
    const int lane = threadIdx.x & 31;
    const int wave = threadIdx.x >> 5;
    const long w = (long)blockIdx.x * (blockDim.x >> 5) + wave;

    const int head = (int)(w % N_HEADS);
    const long nq  = w / N_HEADS;             // n*Lq + q (grid sized exactly)
    const int  n   = (int)(nq / Lq);

    const float* offq = off + (size_t)nq * EMBED_DIM
                            + head * (N_LEVELS * N_POINTS * 2);
    const float* awq  = aw + (size_t)nq * (N_HEADS * N_LEVELS * N_POINTS)
                           + head * (N_LEVELS * N_POINTS);
    const float* rpq  = refp + (size_t)nq * (N_LEVELS * 4);
    const float* vb   = val + (size_t)n * Lv * EMBED_DIM + head * HEAD_DIM + lane;

    static constexpr int Hs[4]   = {100, 50, 25, 13};
    static constexpr int Ws[4]   = {100, 50, 25, 13};
    static constexpr int Base[4] = {0, 10000, 12500, 13125};

    float acc = 0.0f;
#pragma unroll
    for (int lvl = 0; lvl < N_LEVELS; ++lvl) {
        const int   Hl = Hs[lvl], Wd = Ws[lvl], base = Base[lvl];
        const float cx = rpq[lvl * 4 + 0], cy = rpq[lvl * 4 + 1];
        const float rw = rpq[lvl * 4 + 2], rh = rpq[lvl * 4 + 3];
#pragma unroll
        for (int pt = 0; pt < N_POINTS; ++pt) {
            const int pidx = lvl * N_POINTS + pt;
            const float ox = offq[pidx * 2 + 0];
            const float oy = offq[pidx * 2 + 1];
            // loc = cx + off/P * wh * 0.5 ; grid=2*loc-1 ;
            // x = (grid+1)*W/2 - 0.5 = loc*W - 0.5
            const float x = (cx + ox * (0.5f / N_POINTS) * rw) * (float)Wd - 0.5f;
            const float y = (cy + oy * (0.5f / N_POINTS) * rh) * (float)Hl - 0.5f;
            const float x0f = floorf(x), y0f = floorf(y);
            const int   x0 = (int)x0f,  y0 = (int)y0f;
            const float wx1 = x - x0f,  wy1 = y - y0f;
            const float wx0 = 1.0f - wx1, wy0 = 1.0f - wy1;
            const float a = awq[pidx];

            const bool vx0 = (x0 >= 0)     && (x0 < Wd);
            const bool vx1 = (x0 + 1 >= 0) && (x0 + 1 < Wd);
            const bool vy0 = (y0 >= 0)     && (y0 < Hl);
            const bool vy1 = (y0 + 1 >= 0) && (y0 + 1 < Hl);

            float s = 0.0f;
            if (vx0 && vy0)
                s += wx0 * wy0 * vb[(size_t)(base + y0 * Wd + x0) * EMBED_DIM];
            if (vx1 && vy0)
                s += wx1 * wy0 * vb[(size_t)(base + y0 * Wd + x0 + 1) * EMBED_DIM];
            if (vx0 && vy1)
                s += wx0 * wy1 * vb[(size_t)(base + (y0 + 1) * Wd + x0) * EMBED_DIM];
            if (vx1 && vy1)
                s += wx1 * wy1 * vb[(size_t)(base + (y0 + 1) * Wd + x0 + 1) * EMBED_DIM];
            acc += a * s;
        }
    }
    out[(size_t)nq * EMBED_DIM + head * HEAD_DIM + lane] = acc;
}

// ---------------------------------------------------------------------------
extern "C" void kernel_launch(void* const* d_in, const int* in_sizes, int n_in,
                              void* d_out, int out_size, void* d_ws, size_t ws_size,
                              hipStream_t stream)
{
    const float* query  = (const float*)d_in[0];
    const float* refp   = (const float*)d_in[1];
    const float* value  = (const float*)d_in[2];
    const float* W_off  = (const float*)d_in[3];
    const float* b_off  = (const float*)d_in[4];
    const float* W_attn = (const float*)d_in[5];
    const float* b_attn = (const float*)d_in[6];
    const float* W_val  = (const float*)d_in[7];
    const float* b_val  = (const float*)d_in[8];
    const float* W_out  = (const float*)d_in[9];
    const float* b_out  = (const float*)d_in[10];
    float* out = (float*)d_out;

    const int MQ = in_sizes[0] / EMBED_DIM;   // N * Lq
    const int MV = in_sizes[2] / EMBED_DIM;   // N * Lv
    const int Lv = LV_TOTAL;
    const int Nb = MV / Lv;
    const int Lq = MQ / (Nb > 0 ? Nb : 1);

    // Workspace layout (floats): ~95 MB total.
    float* ws      = (float*)d_ws;
    float* ws_val  = ws;                                            // MV*256
    float* ws_off  = ws_val + (size_t)MV * EMBED_DIM;               // MQ*256
    float* ws_aw   = ws_off + (size_t)MQ * EMBED_DIM;               // MQ*128
    float* ws_att  = ws_aw  + (size_t)MQ * (N_HEADS * N_LEVELS * N_POINTS); // MQ*256

    const int wavesPerBlock = 8;   // 256 threads
    auto nblk = [](long waves, int wpb) { return (int)((waves + wpb - 1) / wpb); };

    // 1) value projection: (MV,256) @ (256,256)
    {
        long tiles = (long)((MV + 31) / 32) * (EMBED_DIM / 64);
        gemm_bias_wmma<0><<<nblk(tiles, wavesPerBlock), 256, 0, stream>>>(
            value, W_val, b_val, ws_val, MV, EMBED_DIM, EMBED_DIM);
    }
    // 2) sampling offsets: (MQ,256) @ (256,256)
    {
        long tiles = (long)((MQ + 31) / 32) * (EMBED_DIM / 64);
        gemm_bias_wmma<0><<<nblk(tiles, wavesPerBlock), 256, 0, stream>>>(
            query, W_off, b_off, ws_off, MQ, EMBED_DIM, EMBED_DIM);
    }
    // 3) attention weights + fused softmax: (MQ,256) @ (256,128)
    {
        const int Nn = N_HEADS * N_LEVELS * N_POINTS;  // 128
        long tiles = (long)((MQ + 31) / 32) * (Nn / 64);
        gemm_bias_wmma<1><<<nblk(tiles, wavesPerBlock), 256, 0, stream>>>(
            query, W_attn, b_attn, ws_aw, MQ, EMBED_DIM, Nn);
    }
    // 4) deformable sampling core: one wave per (n,q,head)
    {
        long waves = (long)MQ * N_HEADS;
        msda_sample_kernel<<<nblk(waves, wavesPerBlock), 256, 0, stream>>>(
            ws_val, ws_off, ws_aw, refp, ws_att, Lq, Lv);
    }
    // 5) output projection: (MQ,256) @ (256,256) -> d_out
    {
        long tiles = (long)((MQ + 31) / 32) * (EMBED_DIM / 64);
        gemm_bias_wmma<0><<<nblk(tiles, wavesPerBlock), 256, 0, stream>>>(
            ws_att, W_out, b_out, out, MQ, EMBED_DIM, EMBED_DIM);
    }
}